// MMDLoss_29618094474227
// MI455X (gfx1250) — compile-verified
//
#include <hip/hip_runtime.h>
#include <hip/hip_bf16.h>
#include <math.h>

typedef __attribute__((ext_vector_type(2))) float v2f;
typedef __attribute__((ext_vector_type(4))) float v4f;
typedef __attribute__((ext_vector_type(8))) float v8f;

#define BS   4096   // rows per input
#define NTOT 8192   // concat rows
#define D    256    // feature dim
#define BT   64     // gram block tile (BT x BT per workgroup)
#define KC   64     // K chunk staged in LDS
#define LDP  68     // padded LDS row stride (floats): (4*row + k) % 64 conflict-free

__device__ __forceinline__ const float* row_ptr(const float* __restrict__ s,
                                                const float* __restrict__ t, int r) {
  return (r < BS) ? (s + (size_t)r * D) : (t + (size_t)(r - BS) * D);
}

// ---------------- kernel 1: per-row squared norms (one wave per row) --------
__global__ __launch_bounds__(256)
void sq_kernel(const float* __restrict__ src, const float* __restrict__ tgt,
               float* __restrict__ sq) {
  const int wave = threadIdx.x >> 5;
  const int lane = threadIdx.x & 31;
  const int row  = blockIdx.x * 8 + wave;
  const float* p = row_ptr(src, tgt, row);
  float acc = 0.f;
#pragma unroll
  for (int i = 0; i < D; i += 32) {
    float v = p[i + lane];
    acc += v * v;
  }
#pragma unroll
  for (int m = 16; m; m >>= 1) acc += __shfl_xor(acc, m, 32);
  if (lane == 0) sq[row] = acc;
}

// ---------------- kernel 2: bandwidth scalar -------------------------------
// sum(L2) = 2*n*sum(sq) - 2*||colsum||^2   (no Gram needed)
__global__ __launch_bounds__(256)
void band_kernel(const float* __restrict__ src, const float* __restrict__ tgt,
                 const float* __restrict__ sq, float* __restrict__ inv_b16) {
  __shared__ float redA[256];
  __shared__ float redB[256];
  const int t = threadIdx.x;
  float s = 0.f;
  for (int r = 0; r < NTOT; ++r) s += row_ptr(src, tgt, r)[t];
  float ss = 0.f;
  for (int i = t; i < NTOT; i += 256) ss += sq[i];
  redA[t] = s * s;
  redB[t] = ss;
  __syncthreads();
  for (int k = 128; k; k >>= 1) {
    if (t < k) { redA[t] += redA[t + k]; redB[t] += redB[t + k]; }
    __syncthreads();
  }
  if (t == 0) {
    double n      = (double)NTOT;
    double sumL2  = 2.0 * n * (double)redB[0] - 2.0 * (double)redA[0];
    double bw     = sumL2 / (n * n - n);
    bw            = bw / 4.0;               // KERNEL_MUL^(KERNEL_NUM/2) = 2^2
    inv_b16[0]    = (float)(1.0 / (bw * 16.0));
  }
}

// ---------------- kernel 3: WMMA Gram tile + exp epilogue ------------------
__global__ __launch_bounds__(256)
void mmd_kernel(const float* __restrict__ src, const float* __restrict__ tgt,
                const float* __restrict__ sq, const float* __restrict__ inv_b16p,
                float* __restrict__ partials) {
  __shared__ float lA[BT * LDP];
  __shared__ float lB[BT * LDP];
  __shared__ float red[256];

  const int mbase = blockIdx.y * BT;
  const int nbase = blockIdx.x * BT;
  const int tid   = threadIdx.x;
  const int lane  = tid & 31;
  const int wave  = tid >> 5;
  const int mt    = wave & 3;          // M-tile 0..3 inside block
  const int ntp   = wave >> 2;         // 0/1 -> N-tiles {0,1} or {2,3}

  v8f acc0 = {};
  v8f acc1 = {};

  const int rsel  = (lane >> 4) << 1;          // 0 or 2 (K sub-pair)
  const int mrow  = mt * 16 + (lane & 15);
  const int n0row = (2 * ntp) * 16 + (lane & 15);
  const int n1row = n0row + 16;

  for (int kc = 0; kc < D; kc += KC) {
    // stage 64x64 chunks of A-rows and B-rows into LDS (float4, padded stride)
#pragma unroll
    for (int i = 0; i < 4; ++i) {
      int idx = tid + i * 256;         // 0..1023 float4 slots
      int row = idx >> 4;              // 0..63
      int col = (idx & 15) << 2;       // 0..60
      v4f va = *(const v4f*)(row_ptr(src, tgt, mbase + row) + kc + col);
      *(v4f*)&lA[row * LDP + col] = va;
      v4f vb = *(const v4f*)(row_ptr(src, tgt, nbase + row) + kc + col);
      *(v4f*)&lB[row * LDP + col] = vb;
    }
    __syncthreads();
#pragma unroll
    for (int k = 0; k < KC; k += 4) {
      v2f a  = *(const v2f*)&lA[mrow  * LDP + k + rsel];
      v2f b0 = *(const v2f*)&lB[n0row * LDP + k + rsel];
      v2f b1 = *(const v2f*)&lB[n1row * LDP + k + rsel];
      acc0 = __builtin_amdgcn_wmma_f32_16x16x4_f32(false, a, false, b0,
                                                   (short)0, acc0, false, false);
      acc1 = __builtin_amdgcn_wmma_f32_16x16x4_f32(false, a, false, b1,
                                                   (short)0, acc1, false, false);
    }
    __syncthreads();
  }

  // epilogue: L2 -> u = exp(-L2/(16b)); kernels = u + u^2 + u^4 + u^8 + u^16
  const float ib       = inv_b16p[0];
  const int   gRowBase = mbase + mt * 16 + ((lane >> 4) << 3);
  const int   gCol0    = nbase + (2 * ntp) * 16 + (lane & 15);
  const int   gCol1    = gCol0 + 16;
  const float sqc0     = sq[gCol0];
  const float sqc1     = sq[gCol1];
  const bool  c0src    = gCol0 < BS;
  const bool  c1src    = gCol1 < BS;

  float accv = 0.f;
#pragma unroll
  for (int v = 0; v < 8; ++v) {
    const int   gRow = gRowBase + v;
    const float sr   = sq[gRow];
    const bool  rsrc = gRow < BS;
    {
      float L2 = sr + sqc0 - 2.f * acc0[v];
      float u  = __expf(-L2 * ib);
      float u2 = u * u, u4 = u2 * u2, u8 = u4 * u4, u16 = u8 * u8;
      float kv = u + u2 + u4 + u8 + u16;
      accv += (rsrc == c0src) ? kv : -kv;
    }
    {
      float L2 = sr + sqc1 - 2.f * acc1[v];
      float u  = __expf(-L2 * ib);
      float u2 = u * u, u4 = u2 * u2, u8 = u4 * u4, u16 = u8 * u8;
      float kv = u + u2 + u4 + u8 + u16;
      accv += (rsrc == c1src) ? kv : -kv;
    }
  }

  red[tid] = accv;
  __syncthreads();
  for (int k = 128; k; k >>= 1) {
    if (tid < k) red[tid] += red[tid + k];
    __syncthreads();
  }
  if (tid == 0) partials[blockIdx.y * gridDim.x + blockIdx.x] = red[0];
}

// ---------------- kernel 4: deterministic final reduce ---------------------
__global__ __launch_bounds__(256)
void reduce_kernel(const float* __restrict__ partials, int n, float* __restrict__ out) {
  __shared__ float red[256];
  const int t = threadIdx.x;
  float s = 0.f;
  for (int i = t; i < n; i += 256) s += partials[i];
  red[t] = s;
  __syncthreads();
  for (int k = 128; k; k >>= 1) {
    if (t < k) red[t] += red[t + k];
    __syncthreads();
  }
  if (t == 0) out[0] = red[0] * (1.0f / ((float)BS * (float)BS));
}

extern "C" void kernel_launch(void* const* d_in, const int* in_sizes, int n_in,
                              void* d_out, int out_size, void* d_ws, size_t ws_size,
                              hipStream_t stream) {
  const float* src = (const float*)d_in[0];
  const float* tgt = (const float*)d_in[1];
  float* ws       = (float*)d_ws;
  float* sq       = ws;                 // NTOT floats
  float* ib       = ws + NTOT;          // 1 float (+pad)
  float* partials = ws + NTOT + 256;    // (NTOT/BT)^2 = 16384 floats

  sq_kernel<<<NTOT / 8, 256, 0, stream>>>(src, tgt, sq);
  band_kernel<<<1, 256, 0, stream>>>(src, tgt, sq, ib);
  dim3 grid(NTOT / BT, NTOT / BT);
  mmd_kernel<<<grid, 256, 0, stream>>>(src, tgt, sq, ib, partials);
  reduce_kernel<<<1, 256, 0, stream>>>(partials, (NTOT / BT) * (NTOT / BT),
                                       (float*)d_out);
}